// Aggregator_59811714564548
// MI455X (gfx1250) — compile-verified
//
#include <hip/hip_runtime.h>
#include <hip/hip_bf16.h>
#include <math.h>

#define N_ENTITIES 100000
#define N_ITEMS    20000
#define N_USERS    50000
#define DD         64
#define EPSF       1e-15f

typedef __attribute__((ext_vector_type(2))) float v2f;
typedef __attribute__((ext_vector_type(8))) float v8f;

// ---------- wave32 helpers ----------
__device__ __forceinline__ float wred(float v) {
    // full-wave (32-lane) sum reduction; result uniform across lanes
    v += __shfl_xor(v, 16, 32);
    v += __shfl_xor(v, 8, 32);
    v += __shfl_xor(v, 4, 32);
    v += __shfl_xor(v, 2, 32);
    v += __shfl_xor(v, 1, 32);
    return v;
}

__device__ __forceinline__ v8f wmma4(v2f a, v2f b, v8f c) {
    // V_WMMA_F32_16X16X4_F32 : D = A(16x4 f32) x B(4x16 f32) + C(16x16 f32)
    return __builtin_amdgcn_wmma_f32_16x16x4_f32(
        /*neg_a=*/false, a, /*neg_b=*/false, b,
        /*c_mod=*/(short)0, c, /*reuse_a=*/false, /*reuse_b=*/false);
}

// ---------- zero workspace ----------
__global__ void zero_kernel(float* __restrict__ p, size_t n) {
    size_t i = (size_t)blockIdx.x * blockDim.x + threadIdx.x;
    size_t st = (size_t)gridDim.x * blockDim.x;
    for (; i < n; i += st) p[i] = 0.0f;
}

// ---------- expmap(u, p) followed by mobius_add(p, .) (per-lane float2 slice) ----------
__device__ __forceinline__ v2f expmap_at_p(v2f u, v2f p, float p2, float lam) {
    float nu2 = wred(u.x * u.x + u.y * u.y);
    float nu  = fmaxf(sqrtf(nu2), EPSF);
    float s   = tanhf(0.5f * lam * nu);
    float k   = s / nu;
    v2f sec; sec.x = u.x * k; sec.y = u.y * k;
    float y2 = s * s;                       // ||second||^2 closed form
    float xy = wred(p.x * sec.x + p.y * sec.y);
    float cx = 1.0f + 2.0f * xy + y2;
    float cy = 1.0f - p2;
    float inv = 1.0f / fmaxf(1.0f + 2.0f * xy + p2 * y2, EPSF);
    v2f r;
    r.x = (cx * p.x + cy * sec.x) * inv;
    r.y = (cx * p.y + cy * sec.y) * inv;
    return r;
}

// ---------- per-edge hyperbolic transform + scatter-mean ----------
__global__ void edge_kernel(const float* __restrict__ ent,
                            const float* __restrict__ rel_emb,
                            const int* __restrict__ esrc,
                            const int* __restrict__ edst,
                            const int* __restrict__ etyp,
                            float* __restrict__ sum_out,
                            float* __restrict__ cnt_out,
                            int n_edges) {
    int wave = (int)((blockIdx.x * (size_t)blockDim.x + threadIdx.x) >> 5);
    int lane = threadIdx.x & 31;
    if (wave >= n_edges) return;

    int s = esrc[wave];
    int d = edst[wave];
    int t = etyp[wave] + 2;

    const v2f* es2 = (const v2f*)(ent + (size_t)s * DD);
    const v2f* ed2 = (const v2f*)(ent + (size_t)d * DD);
    const v2f* rr2 = (const v2f*)(rel_emb + (size_t)t * DD);
    v2f es = es2[lane];
    v2f ed = ed2[lane];
    v2f rr = rr2[lane];

    // p = expmap0(e_src)
    float ne2 = wred(es.x * es.x + es.y * es.y);
    float ne  = fmaxf(sqrtf(ne2), EPSF);
    float th  = tanhf(ne);
    float sc  = th / ne;
    v2f p; p.x = es.x * sc; p.y = es.y * sc;
    float p2  = th * th;
    float lam = 2.0f / fmaxf(1.0f - p2, EPSF);

    // a = expmap(dst, p), b = expmap(rel, p)
    v2f a = expmap_at_p(ed, p, p2, lam);
    v2f b = expmap_at_p(rr, p, p2, lam);

    // c = mobius_add(a, b)
    float a2 = wred(a.x * a.x + a.y * a.y);
    float b2 = wred(b.x * b.x + b.y * b.y);
    float ab = wred(a.x * b.x + a.y * b.y);
    float cxa = 1.0f + 2.0f * ab + b2;
    float cyb = 1.0f - a2;
    float inv = 1.0f / fmaxf(1.0f + 2.0f * ab + a2 * b2, EPSF);
    v2f c;
    c.x = (cxa * a.x + cyb * b.x) * inv;
    c.y = (cxa * a.y + cyb * b.y) * inv;

    // project(c)
    float c2 = wred(c.x * c.x + c.y * c.y);
    float nc = fmaxf(sqrtf(c2), EPSF);
    const float maxn = 1.0f - 1e-5f;
    if (nc > maxn) {            // wave-uniform branch (nc reduced)
        float ps = maxn / nc;
        c.x *= ps; c.y *= ps;
        c2 = maxn * maxn;
    }

    // logmap(c, p): sub = mobius_add(-p, c)
    float pc = wred(p.x * c.x + p.y * c.y);
    float xy = -pc;
    float nx = 1.0f + 2.0f * xy + c2;   // coeff on x = -p
    float ny = 1.0f - p2;
    float inv2 = 1.0f / fmaxf(1.0f + 2.0f * xy + p2 * c2, EPSF);
    v2f sub;
    sub.x = (nx * (-p.x) + ny * c.x) * inv2;
    sub.y = (nx * (-p.y) + ny * c.y) * inv2;
    float s2 = wred(sub.x * sub.x + sub.y * sub.y);
    float sn = fmaxf(sqrtf(s2), EPSF);
    float z  = fminf(sn, 1.0f - 1e-7f);
    float at = 0.5f * (log1pf(z) - log1pf(-z));
    float fac = (2.0f / lam) * at / sn;

    float* dst0 = sum_out + (size_t)s * DD + 2 * lane;
    atomicAdd(dst0,     sub.x * fac);
    atomicAdd(dst0 + 1, sub.y * fac);
    if (lane == 0) atomicAdd(&cnt_out[s], 1.0f);
}

// ---------- seg-mean divide + squared norms (entities and users) ----------
__global__ void finalize_out_kernel(const float* __restrict__ sum_out,
                                    const float* __restrict__ cnt_out,
                                    const float* __restrict__ user_embed,
                                    float* __restrict__ d_out,
                                    float* __restrict__ norm_item,
                                    float* __restrict__ norm_user) {
    const size_t SEG2 = (size_t)N_ENTITIES * DD + (size_t)N_USERS * DD;
    int wave = (int)((blockIdx.x * (size_t)blockDim.x + threadIdx.x) >> 5);
    int lane = threadIdx.x & 31;
    if (wave >= N_ENTITIES + N_USERS) return;

    if (wave < N_ENTITIES) {
        float invc = 1.0f / fmaxf(cnt_out[wave], 1.0f);
        const v2f* s2 = (const v2f*)(sum_out + (size_t)wave * DD);
        v2f v = s2[lane];
        v.x *= invc; v.y *= invc;
        if (wave < N_ITEMS) {
            v2f* o = (v2f*)(d_out + SEG2 + (size_t)wave * DD);
            o[lane] = v;
            float n2 = wred(v.x * v.x + v.y * v.y);
            if (lane == 0) norm_item[wave] = n2;
        } else {
            v2f* o = (v2f*)(d_out + (size_t)wave * DD);
            o[lane] = v;
        }
    } else {
        int u = wave - N_ENTITIES;
        const v2f* s2 = (const v2f*)(user_embed + (size_t)u * DD);
        v2f v = s2[lane];
        float n2 = wred(v.x * v.x + v.y * v.y);
        if (lane == 0) norm_user[u] = n2;
    }
}

// ---------- bipartite interaction scatter (both directions) ----------
__global__ void inter_kernel(const float* __restrict__ d_out,      // seg2 = out_items
                             const float* __restrict__ user_embed,
                             const int* __restrict__ iitem,
                             const int* __restrict__ iuser,
                             const float* __restrict__ norm_item,
                             const float* __restrict__ norm_user,
                             float* __restrict__ sum_u,
                             float* __restrict__ cnt_u,
                             float* __restrict__ sum_icf,
                             float* __restrict__ cnt_icf,
                             int n_inter) {
    const size_t SEG2 = (size_t)N_ENTITIES * DD + (size_t)N_USERS * DD;
    int wave = (int)((blockIdx.x * (size_t)blockDim.x + threadIdx.x) >> 5);
    int lane = threadIdx.x & 31;
    if (wave >= n_inter) return;

    int it = iitem[wave];
    int uu = iuser[wave];

    // g_i2u: copy item row onto user mean
    const v2f* xi2 = (const v2f*)(d_out + SEG2 + (size_t)it * DD);
    v2f xi = xi2[lane];
    float* su = sum_u + (size_t)uu * DD + 2 * lane;
    atomicAdd(su,     xi.x);
    atomicAdd(su + 1, xi.y);

    // g_u2i: (user / (||user||^2 + 1e-6)) * ||item||^2 onto item mean
    float s = norm_item[it] / (norm_user[uu] + 1e-6f);
    const v2f* ue2 = (const v2f*)(user_embed + (size_t)uu * DD);
    v2f ue = ue2[lane];
    float* si = sum_icf + (size_t)it * DD + 2 * lane;
    atomicAdd(si,     ue.x * s);
    atomicAdd(si + 1, ue.y * s);

    if (lane == 0) {
        atomicAdd(&cnt_u[uu], 1.0f);
        atomicAdd(&cnt_icf[it], 1.0f);
    }
}

// ---------- divide u (-> d_out seg1) and i_cf (in place) ----------
__global__ void divide_kernel(const float* __restrict__ sum_u,
                              const float* __restrict__ cnt_u,
                              float* __restrict__ sum_icf,
                              const float* __restrict__ cnt_icf,
                              float* __restrict__ d_out) {
    const size_t SEG1 = (size_t)N_ENTITIES * DD;
    int wave = (int)((blockIdx.x * (size_t)blockDim.x + threadIdx.x) >> 5);
    int lane = threadIdx.x & 31;
    if (wave >= N_USERS + N_ITEMS) return;

    if (wave < N_USERS) {
        float invc = 1.0f / fmaxf(cnt_u[wave], 1.0f);
        const v2f* s2 = (const v2f*)(sum_u + (size_t)wave * DD);
        v2f v = s2[lane];
        v.x *= invc; v.y *= invc;
        v2f* o = (v2f*)(d_out + SEG1 + (size_t)wave * DD);
        o[lane] = v;
    } else {
        int it = wave - N_USERS;
        float invc = 1.0f / fmaxf(cnt_icf[it], 1.0f);
        v2f* s2 = (v2f*)(sum_icf + (size_t)it * DD);
        v2f v = s2[lane];
        v.x *= invc; v.y *= invc;
        s2[lane] = v;
    }
}

// ---------- gated fusion with f32 WMMA: gi = sigmoid(X@W1^T + icf@W2^T) ----------
__global__ void fusion_wmma_kernel(const float* __restrict__ d_out_ro, // seg2 = X (out_items)
                                   const float* __restrict__ icf,
                                   const float* __restrict__ W1,
                                   const float* __restrict__ W2,
                                   float* __restrict__ d_out) {
    const size_t SEG2 = (size_t)N_ENTITIES * DD + (size_t)N_USERS * DD;
    const int NTILES = N_ITEMS / 16;   // 1250
    int wave = (int)((blockIdx.x * (size_t)blockDim.x + threadIdx.x) >> 5);
    int lane = threadIdx.x & 31;
    if (wave >= NTILES) return;        // wave-uniform exit keeps EXEC all-1

    const float* X = d_out_ro + SEG2;
    int rowBase = wave * 16;
    int hi  = (lane >= 16) ? 1 : 0;    // lane group selects K pair
    int l16 = lane & 15;

    v8f acc[4];
    acc[0] = (v8f)0.0f; acc[1] = (v8f)0.0f; acc[2] = (v8f)0.0f; acc[3] = (v8f)0.0f;

    // A fragment (16x4 f32): lane<16 holds A[m][k0..k0+1], lane>=16 holds A[m][k0+2..k0+3]
    int mA = rowBase + l16;
    for (int kk = 0; kk < 16; ++kk) {
        int k0 = kk * 4 + 2 * hi;
        v2f a1 = *(const v2f*)(X   + (size_t)mA * DD + k0);
        v2f a2 = *(const v2f*)(icf + (size_t)mA * DD + k0);
        #pragma unroll
        for (int nt = 0; nt < 4; ++nt) {
            int n = nt * 16 + l16;
            // B fragment (4x16 f32): Bmat[k][n] = W[n][k] (result = X @ W^T)
            v2f b1 = *(const v2f*)(W1 + (size_t)n * DD + k0);
            v2f b2 = *(const v2f*)(W2 + (size_t)n * DD + k0);
            acc[nt] = wmma4(a1, b1, acc[nt]);
            acc[nt] = wmma4(a2, b2, acc[nt]);
        }
    }

    // epilogue: D layout — vgpr r: lanes 0-15 -> M=r, lanes 16-31 -> M=r+8; N = lane%16
    #pragma unroll
    for (int nt = 0; nt < 4; ++nt) {
        int col = nt * 16 + l16;
        #pragma unroll
        for (int r = 0; r < 8; ++r) {
            int m = rowBase + r + 8 * hi;
            float g  = 1.0f / (1.0f + __expf(-acc[nt][r]));
            float xo = X[(size_t)m * DD + col];
            float ic = icf[(size_t)m * DD + col];
            d_out[(size_t)m * DD + col] = g * xo + (1.0f - g) * ic;
        }
    }
}

extern "C" void kernel_launch(void* const* d_in, const int* in_sizes, int n_in,
                              void* d_out, int out_size, void* d_ws, size_t ws_size,
                              hipStream_t stream) {
    const float* ent   = (const float*)d_in[0];
    const float* usr   = (const float*)d_in[1];
    const float* relE  = (const float*)d_in[2];
    const float* W1    = (const float*)d_in[3];
    const float* W2    = (const float*)d_in[4];
    const int*   esrc  = (const int*)d_in[5];
    const int*   edst  = (const int*)d_in[6];
    const int*   etyp  = (const int*)d_in[7];
    const int*   iitem = (const int*)d_in[8];
    const int*   iuser = (const int*)d_in[9];
    float* out = (float*)d_out;
    float* ws  = (float*)d_ws;

    const int n_edges = in_sizes[5];
    const int n_inter = in_sizes[8];

    // workspace layout (floats)
    float* sum_out   = ws;                                      // 6,400,000
    float* cnt_out   = sum_out + (size_t)N_ENTITIES * DD;       //   100,000
    float* sum_u     = cnt_out + N_ENTITIES;                    // 3,200,000
    float* cnt_u     = sum_u + (size_t)N_USERS * DD;            //    50,000
    float* sum_icf   = cnt_u + N_USERS;                         // 1,280,000
    float* cnt_icf   = sum_icf + (size_t)N_ITEMS * DD;          //    20,000
    float* norm_item = cnt_icf + N_ITEMS;                       //    20,000
    float* norm_user = norm_item + N_ITEMS;                     //    50,000
    size_t ws_floats = (size_t)(norm_user + N_USERS - ws);

    zero_kernel<<<2048, 256, 0, stream>>>(ws, ws_floats);

    edge_kernel<<<(n_edges + 7) / 8, 256, 0, stream>>>(
        ent, relE, esrc, edst, etyp, sum_out, cnt_out, n_edges);

    finalize_out_kernel<<<(N_ENTITIES + N_USERS + 7) / 8, 256, 0, stream>>>(
        sum_out, cnt_out, usr, out, norm_item, norm_user);

    inter_kernel<<<(n_inter + 7) / 8, 256, 0, stream>>>(
        out, usr, iitem, iuser, norm_item, norm_user,
        sum_u, cnt_u, sum_icf, cnt_icf, n_inter);

    divide_kernel<<<(N_USERS + N_ITEMS + 7) / 8, 256, 0, stream>>>(
        sum_u, cnt_u, sum_icf, cnt_icf, out);

    fusion_wmma_kernel<<<(N_ITEMS / 16 + 7) / 8, 256, 0, stream>>>(
        out, sum_icf, W1, W2, out);
}